// DihedralAngleEncoder_23003844837869
// MI455X (gfx1250) — compile-verified
//
#include <hip/hip_runtime.h>

// Fused dihedral-angle encoder for MI455X (gfx1250, wave32).
// Memory-bound kernel (~80 MB traffic -> ~3.4 us floor @ 23.3 TB/s).
// sin/cos(arccos) collapse algebraically -> no transcendentals.
// 6->64 projection done with V_WMMA_F32_16X16X4_F32 (K padded 6->8).

#define EPSV 1e-8f
#define LSEQ 4096
#define DOUT 64

typedef float v2f __attribute__((ext_vector_type(2)));
typedef float v8f __attribute__((ext_vector_type(8)));

struct F3 { float x, y, z; };

__device__ __forceinline__ F3 sub3(F3 a, F3 b) { return {a.x - b.x, a.y - b.y, a.z - b.z}; }
__device__ __forceinline__ F3 cross3(F3 a, F3 b) {
    return {a.y * b.z - a.z * b.y, a.z * b.x - a.x * b.z, a.x * b.y - a.y * b.x};
}
__device__ __forceinline__ float dot3(F3 a, F3 b) { return a.x * b.x + a.y * b.y + a.z * b.z; }

// Returns (sin, cos) of the signed dihedral without arccos/sin/cos:
// angle = arccos(c) * sign  =>  cos(angle) = c (sign!=0), sin(angle) = sign*sqrt(1-c^2).
__device__ __forceinline__ void dihedral_sc(F3 p1, F3 p2, F3 p3, F3 p4, float& s, float& c) {
    F3 v1 = sub3(p2, p1), v2 = sub3(p3, p2), v3 = sub3(p4, p3);
    F3 n1 = cross3(v1, v2), n2 = cross3(v2, v3);
    float inv1 = 1.0f / (sqrtf(dot3(n1, n1)) + EPSV);
    float inv2 = 1.0f / (sqrtf(dot3(n2, n2)) + EPSV);
    F3 n1n = {n1.x * inv1, n1.y * inv1, n1.z * inv1};
    F3 n2n = {n2.x * inv2, n2.y * inv2, n2.z * inv2};
    float ca = dot3(n1n, n2n);
    ca = fminf(1.0f, fmaxf(-1.0f, ca));
    float sd = dot3(n1n, v3);
    float sg = (sd > 0.0f) ? 1.0f : ((sd < 0.0f) ? -1.0f : 0.0f);
    s = sg * sqrtf(fmaxf(1.0f - ca * ca, 0.0f));
    c = (sg == 0.0f) ? 1.0f : ca;   // sign==0 -> angle==0 -> (sin,cos)=(0,1)
}

__global__ __launch_bounds__(256)
void dihedral_encode_wmma(const float* __restrict__ coords,
                          const float* __restrict__ W,
                          const float* __restrict__ bias,
                          float* __restrict__ out) {
    __shared__ float shC[258 * 12];   // residues l0-1 .. l0+256 (N,CA,C,O xyz)
    __shared__ float shE[256 * 9];    // per-row features, stride 9 (bank-conflict-free)
    __shared__ float shW[8 * 64];     // padded W^T: shW[k][n] = W[n][k], k=6,7 zero
    __shared__ float shB[64];         // bias

    const int tid = threadIdx.x;
    const int blockRow0 = blockIdx.x * 256;      // global row (b*L + l) start
    const int batch = blockRow0 / LSEQ;          // 256 | 4096 -> block stays in one batch
    const int l0 = blockRow0 % LSEQ;

    // ---- Stage coords (coalesced) ----
    const float* cbase = coords + (size_t)batch * LSEQ * 12;
    for (int i = tid; i < 258 * 12; i += 256) {
        int ri = i / 12;
        int cc = i - ri * 12;
        int lg = l0 - 1 + ri;
        lg = lg < 0 ? 0 : (lg > LSEQ - 1 ? LSEQ - 1 : lg);
        shC[i] = cbase[lg * 12 + cc];
    }
    // ---- Stage padded W^T and bias ----
    for (int i = tid; i < 8 * 64; i += 256) {
        int k = i >> 6;
        int n = i & 63;
        shW[i] = (k < 6) ? W[n * 6 + k] : 0.0f;
    }
    if (tid < 64) shB[tid] = bias[tid];
    __syncthreads();

    // ---- Per-lane geometry: one row each ----
    {
        const int li = tid;
        const int l = l0 + li;
        const float* r0 = &shC[(li) * 12];       // residue l-1 (clamped)
        const float* r1 = &shC[(li + 1) * 12];   // residue l
        const float* r2 = &shC[(li + 2) * 12];   // residue l+1 (clamped)
        F3 Cp  = {r0[6], r0[7], r0[8]};
        F3 N   = {r1[0], r1[1], r1[2]};
        F3 CA  = {r1[3], r1[4], r1[5]};
        F3 C   = {r1[6], r1[7], r1[8]};
        F3 Nn  = {r2[0], r2[1], r2[2]};
        F3 CAn = {r2[3], r2[4], r2[5]};
        bool last = (l == LSEQ - 1);
        F3 om3 = last ? CA : Nn;    // reference: om_p3 at chain end is CA[L-1]
        F3 om4 = CAn;               // clamp already gives CA[L-1] at chain end
        float s_phi, c_phi, s_psi, c_psi, s_om, c_om;
        dihedral_sc(Cp, N, CA, C, s_phi, c_phi);
        dihedral_sc(N, CA, C, Nn, s_psi, c_psi);
        dihedral_sc(CA, C, om3, om4, s_om, c_om);
        float* e = &shE[li * 9];
        e[0] = s_phi; e[1] = s_psi; e[2] = s_om;
        e[3] = c_phi; e[4] = c_psi; e[5] = c_om;
        e[6] = 0.0f;  e[7] = 0.0f;               // K padding 6->8
    }
    __syncthreads();

    // ---- WMMA projection: per wave, 32 rows x 64 cols ----
    // f32 A 16x4 layout: lane m=lane&15; VGPR0 holds K=2*kh, VGPR1 K=2*kh+1 (kh=lane>>4).
    // B 4x16 mirrored (lane = N). D: VGPR v -> M = v + 8*kh, N = lane&15.
    const int lane = tid & 31;
    const int wv = tid >> 5;
    const int m = lane & 15;
    const int kh = lane >> 4;
    const int wrow = wv * 32;

#pragma unroll
    for (int mt = 0; mt < 2; ++mt) {
        const int arow = (wrow + mt * 16 + m) * 9;
        v2f a0, a1;
        a0.x = shE[arow + 2 * kh + 0];
        a0.y = shE[arow + 2 * kh + 1];
        a1.x = shE[arow + 4 + 2 * kh + 0];
        a1.y = shE[arow + 4 + 2 * kh + 1];
#pragma unroll
        for (int nt = 0; nt < 4; ++nt) {
            const int col = nt * 16 + m;
            v2f b0, b1;
            b0.x = shW[(2 * kh + 0) * 64 + col];
            b0.y = shW[(2 * kh + 1) * 64 + col];
            b1.x = shW[(4 + 2 * kh + 0) * 64 + col];
            b1.y = shW[(4 + 2 * kh + 1) * 64 + col];
            const float bb = shB[col];
            v8f acc = {bb, bb, bb, bb, bb, bb, bb, bb};  // seed C with bias
            acc = __builtin_amdgcn_wmma_f32_16x16x4_f32(
                false, a0, false, b0, (short)0, acc, false, false);
            acc = __builtin_amdgcn_wmma_f32_16x16x4_f32(
                false, a1, false, b1, (short)0, acc, false, false);
            float* op = out + (size_t)(blockRow0 + wrow + mt * 16 + 8 * kh) * DOUT + col;
#pragma unroll
            for (int v = 0; v < 8; ++v) {
                op[(size_t)v * DOUT] = acc[v];
            }
        }
    }
}

extern "C" void kernel_launch(void* const* d_in, const int* in_sizes, int n_in,
                              void* d_out, int out_size, void* d_ws, size_t ws_size,
                              hipStream_t stream) {
    const float* coords = (const float*)d_in[0];  // (B, L, 4, 3) f32
    const float* W      = (const float*)d_in[1];  // (64, 6) f32
    const float* bias   = (const float*)d_in[2];  // (64,) f32
    float* out = (float*)d_out;                   // (B, L, 64) f32

    const int rows = in_sizes[0] / 12;            // B * L
    const int blocks = rows / 256;                // 256 rows per block (L=4096 divisible)
    dihedral_encode_wmma<<<blocks, 256, 0, stream>>>(coords, W, bias, out);
}